// HeightRenderer_40261023432757
// MI455X (gfx1250) — compile-verified
//
#include <hip/hip_runtime.h>

// ---------------------------------------------------------------------------
// HeightRenderer for MI455X (gfx1250, wave32).
//
// depth = sum_{s=0..63} exp(-c[s]) - 64*exp(-c[64]),  c = strict prefix sum of
// the z-lerped density sd[k] (x/y grid coords are exact pixel centers, so only
// z-lerp survives the grid_sample; the rendering integral telescopes).
//
// Hierarchical prefix sum:
//   - 16 rays per wave along N; K split into 4 chunks of 16.
//   - Per chunk q: 4 chained V_WMMA_F32_16X16X4_F32 with the *diagonal*
//     triangular A-tile -> local prefix sums (16 WMMAs total). The 4 A-masks
//     are chunk-independent and built once (4jj + 2hf + t <= laneM).
//   - Inter-chunk carries are rank-1 (all-ones tiles add the chunk sum to
//     every row), so they're applied as per-ray scalars at finalize time:
//     chunksum[q] = acc[q][7] (hf=1 half, m=15), broadcast with a single
//     ds_swizzle SWAPX16 (no lane-index setup, unlike ds_bpermute).
//
// Address math: s = 16q + 4jj + 2hf + t has only two runtime variants (hf).
// Both variants' slice byte-offsets / lerp weight / ground flags fold to
// literals; one v_cndmask blends them per load. The base pointer is
// wave-uniform (one 16-ray tile = same b,h) and forced into SGPRs with
// readfirstlane -> saddr + 32-bit voffset loads. Ground slab is branch-free
// (clamped slice + cndmask; only s=63 touches it).
// ---------------------------------------------------------------------------

typedef __attribute__((ext_vector_type(2))) float v2f;
typedef __attribute__((ext_vector_type(8))) float v8f;

#define BS_    4
#define W_     256
#define H_     256
#define DEPTH_ 64
#define SLICE_ (H_ * W_)          // 65536 floats per z-slice
#define GROUND 1000.0f

// Cross-half-wave swap (lane L <-> lane L^16): ds_swizzle group-of-32 mode,
// offset = xor_mask(0x10) << 10 | or_mask(0) << 5 | and_mask(0x1f) = 0x401F.
__device__ __forceinline__ float swap16(float x) {
  return __int_as_float(__builtin_amdgcn_ds_swizzle(__float_as_int(x), 0x401F));
}

// z-sampling constants for depth sample s (folds to literals after unroll).
__device__ __forceinline__ void zcalc(int s, int& off0, int& off1,
                                      float& wz1, int& g0, int& g1) {
  const float step = -2.0f / 63.0f;              // linspace(1,-1,64) step
  const float gz  = 1.0f + (float)s * step;
  const float iz  = (gz + 1.0f) * 32.0f;         // (gz+1)*0.5*(65-1)
  const float fz0 = floorf(iz);
  wz1 = iz - fz0;
  int z0 = (int)fz0;
  int z0c = z0 < 0 ? 0 : (z0 > DEPTH_ ? DEPTH_ : z0);   // dens space [0,64]
  int z1 = z0 + 1;
  int z1c = z1 < 0 ? 0 : (z1 > DEPTH_ ? DEPTH_ : z1);
  const int sl0 = (z0c > 0 ? z0c : 1) - 1;       // density slice in [0,63]
  const int sl1 = (z1c > 0 ? z1c : 1) - 1;
  off0 = sl0 * (SLICE_ * 4);                     // byte offsets
  off1 = sl1 * (SLICE_ * 4);
  g0 = (z0c == 0);                               // ground-slab flags
  g1 = (z1c == 0);
}

__global__ __launch_bounds__(256)
void height_render_wmma(const float* __restrict__ density, float* __restrict__ out) {
  const int lane  = threadIdx.x & 31;
  const int laneM = lane & 15;          // ray-within-tile == N column == A row M
  const int hf    = lane >> 4;          // half-wave: K sub-pair {0,1} vs {2,3}
  const int wave  = blockIdx.x * (blockDim.x >> 5) + (threadIdx.x >> 5);
  const int ray   = wave * 16 + laneM;  // flat = b*65536 + h*256 + w

  // Wave-uniform base (whole tile shares b and h) -> force into SGPRs.
  const int rayb = wave * 16;
  const unsigned long long ubase =
      (unsigned long long)(const char*)density +
      ((unsigned long long)(rayb >> 16) * (DEPTH_ * SLICE_) +
       (unsigned long long)(rayb & 0xFFFF)) * 4ull;
  const unsigned lo = __builtin_amdgcn_readfirstlane((unsigned)ubase);
  const unsigned hi = __builtin_amdgcn_readfirstlane((unsigned)(ubase >> 32));
  const char* cb = (const char*)(((unsigned long long)hi << 32) | lo);
  const int laneOff = laneM * 4;        // per-lane byte offset within a slice

  // The 4 diagonal A-masks (chunk-independent), built once.
  v2f amask[4];
#pragma unroll
  for (int jj = 0; jj < 4; ++jj) {
    const int kk = 4 * jj + 2 * hf;
    amask[jj][0] = (kk     <= laneM) ? 1.0f : 0.0f;
    amask[jj][1] = (kk + 1 <= laneM) ? 1.0f : 0.0f;
  }

  v8f acc[4] = {v8f{}, v8f{}, v8f{}, v8f{}};

  // jj outer / q inner: consecutive WMMAs hit different accumulators.
#pragma unroll
  for (int jj = 0; jj < 4; ++jj) {
#pragma unroll
    for (int q = 0; q < 4; ++q) {
      v2f bv;
#pragma unroll
      for (int t = 0; t < 2; ++t) {
        const int sA = 16 * q + 4 * jj + t;      // hf = 0 variant (constant)
        const int sB = sA + 2;                   // hf = 1 variant (constant)
        int a0, a1, b0, b1, gA0, gA1, gB0, gB1;
        float wA, wB;
        zcalc(sA, a0, a1, wA, gA0, gA1);         // all fold to literals
        zcalc(sB, b0, b1, wB, gB0, gB1);
        const int   off0 = hf ? b0 : a0;         // one cndmask each
        const int   off1 = hf ? b1 : a1;
        const float wz   = hf ? wB : wA;
        const float v0 = *(const float*)(cb + (laneOff + off0));
        const float v1 = *(const float*)(cb + (laneOff + off1));
        const float d0 = ((hf ? gB0 : gA0) != 0) ? GROUND : v0;  // folds away
        const float d1 = ((hf ? gB1 : gA1) != 0) ? GROUND : v1;  // unless s=63
        bv[t] = d0 + wz * (d1 - d0);             // sigma * intv (intv = 1)
      }
      acc[q] = __builtin_amdgcn_wmma_f32_16x16x4_f32(
          /*neg_a=*/false, amask[jj], /*neg_b=*/false, bv,
          /*c_mod=*/(short)0, acc[q], /*reuse_a=*/false, /*reuse_b=*/false);
    }
  }

  // ---- carries: chunksum[q] = local prefix at m=15 (hf=1 half, VGPR 7) ----
  float cs[4];
#pragma unroll
  for (int q = 0; q < 4; ++q) {
    const float top   = acc[q][7];
    const float other = swap16(top);             // single ds_swizzle SWAPX16
    cs[q] = hf ? top : other;                    // both halves get chunk sum
  }
  float car[4];
  car[0] = 0.0f;
  car[1] = cs[0];
  car[2] = car[1] + cs[1];
  car[3] = car[2] + cs[2];

  // ---- finalize: P = sum exp(-c[s]) for s in 1..63, minus 64*exp(-c[64]) ----
  float P = 0.0f;
#pragma unroll
  for (int q = 0; q < 4; ++q) {
#pragma unroll
    for (int g = 0; g < 8; ++g) {
      const int s_out = 16 * q + g + 8 * hf + 1; // this lane's prefix index
      const float e = __expf(-(acc[q][g] + car[q]));
      P += (s_out == 64) ? (-64.0f * e) : e;
    }
  }
  const float total = P + swap16(P);             // merge the two half-waves
  if (hf == 0) out[ray] = 63.0f - total;         // 64 - depth (c[0] folded in)
}

extern "C" void kernel_launch(void* const* d_in, const int* in_sizes, int n_in,
                              void* d_out, int out_size, void* d_ws, size_t ws_size,
                              hipStream_t stream) {
  const float* density = (const float*)d_in[0];
  float* out = (float*)d_out;

  const int total_rays     = BS_ * H_ * W_;      // 262144
  const int rays_per_block = (256 / 32) * 16;    // 8 waves * 16 rays = 128
  dim3 grid(total_rays / rays_per_block);        // 2048 blocks
  height_render_wmma<<<grid, dim3(256), 0, stream>>>(density, out);
}